// ScoreLoss_45913200394831
// MI455X (gfx1250) — compile-verified
//
#include <hip/hip_runtime.h>
#include <cstdint>
#include <math.h>

#define H 512
#define W 512
#define HW (H * W)
#define NCORN 500
#define CAND_CAP 4096
#define NSORT 4096

#define USE_ASYNC_LDS 1

#if __has_builtin(__builtin_amdgcn_wmma_f32_16x16x4_f32)
#define HAVE_WMMA_F32X4 1
#endif

typedef __attribute__((ext_vector_type(2))) float v2f;
typedef __attribute__((ext_vector_type(8))) float v8f;

// 7-tap Gaussian, sigma=1, normalized (double-precision derived)
__device__ __constant__ float c_g7[7] = {
    0.0044330499f, 0.0540055826f, 0.2420362294f, 0.3990502797f,
    0.2420362294f, 0.0540055826f, 0.0044330499f};

// Compile-time banded matrix for the horizontal blur GEMM:
// BandT[n][r] = g[r-n] if 0 <= r-n <= 6 else 0   (Out = VB[16x24] x Band[24x16])
struct BandTab { alignas(16) float v[16][24]; };
constexpr BandTab make_band() {
  BandTab t{};
  const float g[7] = {0.0044330499f, 0.0540055826f, 0.2420362294f,
                      0.3990502797f, 0.2420362294f, 0.0540055826f,
                      0.0044330499f};
  for (int n = 0; n < 16; ++n)
    for (int r = 0; r < 24; ++r) {
      int d = r - n;
      t.v[n][r] = (d >= 0 && d <= 6) ? g[d] : 0.f;
    }
  return t;
}
__device__ __constant__ BandTab c_band = make_band();

__device__ __forceinline__ int clampi(int v, int lo, int hi) {
  return v < lo ? lo : (v > hi ? hi : v);
}
// jnp.pad 'reflect' (mirror, edge not repeated); offsets <= 3 so one fold is enough
__device__ __forceinline__ int refl(int v, int n) {
  if (v < 0) v = -v;
  if (v >= n) v = 2 * n - 2 - v;
  return v;
}

// ---------------- K0: init counters / accumulator ----------------
__global__ void k_init(double* acc, int* candCnt, int* cornCnt, int B) {
  int t = threadIdx.x;
  if (t == 0) acc[0] = 0.0;
  if (t < B) { candCnt[t] = 0; cornCnt[t] = 0; }
}

// ---------------- K1: RGB -> gray ----------------
__global__ void k_gray(const float* __restrict__ imgs, float* __restrict__ gray,
                       int total) {
  int i = blockIdx.x * blockDim.x + threadIdx.x;
  if (i >= total) return;
  int b = i / HW, p = i - b * HW;
  const float* base = imgs + (size_t)b * 3 * HW + p;
  gray[i] = 0.299f * base[0] + 0.587f * base[HW] + 0.114f * base[2 * HW];
}

// ---------------- K2: fused GFTT response, one wave per 16x16 tile ----------
// LDS staging via CDNA5 async global->LDS DMA; horizontal Gaussian via
// V_WMMA_F32_16X16X4_F32 (Out = VB[16x24] x Band[24x16]).
__global__ __launch_bounds__(32) void k_gftt(const float* __restrict__ gray,
                                             float* __restrict__ resp) {
  __shared__ alignas(16) float sg[24][24];     // gray tile + halo
  __shared__ alignas(16) float sp[3][22][22];  // dx2, dy2, dxy (reflect grid)
  __shared__ alignas(16) float svb[3][16][24]; // vertically blurred, K pad 24
  const int lane = threadIdx.x;
  const int b = blockIdx.z;
  const int ty0 = blockIdx.y * 16;
  const int tx0 = blockIdx.x * 16;
  const float* gimg = gray + (size_t)b * HW;

  // Phase 1: gray -> LDS (async DMA; wide b128 path for interior tiles)
#if USE_ASYNC_LDS
  const bool interior =
      (ty0 >= 4) && (ty0 + 20 <= H) && (tx0 >= 4) && (tx0 + 20 <= W);
  if (interior) {
    for (int i = lane; i < 144; i += 32) {       // 24 rows x 6 float4
      int r = i / 6, c4 = (i % 6) * 4;
      unsigned loff = (unsigned)(uintptr_t)(&sg[r][c4]);
      const float* gp = gimg + (ty0 - 4 + r) * W + (tx0 - 4 + c4);
      asm volatile("global_load_async_to_lds_b128 %0, %1, off"
                   :: "v"(loff), "v"(gp) : "memory");
    }
  } else {
    for (int i = lane; i < 576; i += 32) {
      int r = i / 24, c = i % 24;
      int gy = clampi(ty0 - 4 + r, 0, H - 1);
      int gx = clampi(tx0 - 4 + c, 0, W - 1);
      unsigned loff = (unsigned)(uintptr_t)(&sg[r][c]);
      const float* gp = gimg + gy * W + gx;
      asm volatile("global_load_async_to_lds_b32 %0, %1, off"
                   :: "v"(loff), "v"(gp) : "memory");
    }
  }
#if __has_builtin(__builtin_amdgcn_s_wait_asynccnt)
  __builtin_amdgcn_s_wait_asynccnt(0);
#else
  asm volatile("s_wait_asynccnt 0" ::: "memory");
#endif
#else
  for (int i = lane; i < 576; i += 32) {
    int r = i / 24, c = i % 24;
    int gy = clampi(ty0 - 4 + r, 0, H - 1);
    int gx = clampi(tx0 - 4 + c, 0, W - 1);
    sg[r][c] = gimg[gy * W + gx];
  }
#endif
  __syncthreads();

  // Phase 2: Sobel/8 (edge pad) + structure-tensor products on reflect grid
  for (int i = lane; i < 484; i += 32) {
    int r = i / 22, c = i % 22;
    int gy = refl(ty0 - 3 + r, H);
    int gx = refl(tx0 - 3 + c, W);
    int ym = clampi(gy - 1, 0, H - 1) - (ty0 - 4);
    int y0 = gy - (ty0 - 4);
    int yp = clampi(gy + 1, 0, H - 1) - (ty0 - 4);
    int xm = clampi(gx - 1, 0, W - 1) - (tx0 - 4);
    int x0 = gx - (tx0 - 4);
    int xp = clampi(gx + 1, 0, W - 1) - (tx0 - 4);
    float a00 = sg[ym][xm], a01 = sg[ym][x0], a02 = sg[ym][xp];
    float a10 = sg[y0][xm], a12 = sg[y0][xp];
    float a20 = sg[yp][xm], a21 = sg[yp][x0], a22 = sg[yp][xp];
    float dx = ((a02 + 2.f * a12 + a22) - (a00 + 2.f * a10 + a20)) * 0.125f;
    float dy = ((a20 + 2.f * a21 + a22) - (a00 + 2.f * a01 + a02)) * 0.125f;
    sp[0][r][c] = dx * dx;
    sp[1][r][c] = dy * dy;
    sp[2][r][c] = dx * dy;
  }
  __syncthreads();

  // Phase 3: vertical 7-tap blur -> svb (cols 22,23 zero-padded)
  for (int i = lane; i < 16 * 24; i += 32) {
    int r = i / 24, c = i % 24;
    for (int ch = 0; ch < 3; ++ch) {
      float acc = 0.f;
      if (c < 22) {
#pragma unroll
        for (int k = 0; k < 7; ++k) acc = fmaf(c_g7[k], sp[ch][r + k][c], acc);
      }
      svb[ch][r][c] = acc;
    }
  }
  __syncthreads();

  // Phase 4: horizontal blur as 16x16x4 f32 WMMA chain (banded B matrix).
  // Preload all fragments so the 18 WMMAs issue back-to-back.
  const int m = lane & 15;       // A row == D/B column for this lane
  const int hi = lane >> 4;      // K/M half selector
  const int ko = hi * 2;
  float h0[8], h1[8], h2[8];
#if HAVE_WMMA_F32X4
  v2f Af[3][6], Bf[6];
#pragma unroll
  for (int t = 0; t < 6; ++t) {
    int r0 = 4 * t + ko;                         // even -> 8B aligned
    Bf[t] = *(const v2f*)&c_band.v[m][r0];       // B rows (r0, r0+1), col m
    Af[0][t] = *(const v2f*)&svb[0][m][r0];      // A (M=m, K=r0..r0+1)
    Af[1][t] = *(const v2f*)&svb[1][m][r0];
    Af[2][t] = *(const v2f*)&svb[2][m][r0];
  }
  v8f acc0 = {}, acc1 = {}, acc2 = {};
#pragma unroll
  for (int t = 0; t < 6; ++t) {
    acc0 = __builtin_amdgcn_wmma_f32_16x16x4_f32(false, Af[0][t], false, Bf[t],
                                                 (short)0, acc0, false, false);
    acc1 = __builtin_amdgcn_wmma_f32_16x16x4_f32(false, Af[1][t], false, Bf[t],
                                                 (short)0, acc1, false, false);
    acc2 = __builtin_amdgcn_wmma_f32_16x16x4_f32(false, Af[2][t], false, Bf[t],
                                                 (short)0, acc2, false, false);
  }
#pragma unroll
  for (int v = 0; v < 8; ++v) { h0[v] = acc0[v]; h1[v] = acc1[v]; h2[v] = acc2[v]; }
#else
#pragma unroll
  for (int v = 0; v < 8; ++v) {
    int row = v + 8 * hi;
    float s0 = 0.f, s1 = 0.f, s2 = 0.f;
#pragma unroll
    for (int k = 0; k < 7; ++k) {
      float gk = c_g7[k];
      s0 = fmaf(gk, svb[0][row][m + k], s0);
      s1 = fmaf(gk, svb[1][row][m + k], s1);
      s2 = fmaf(gk, svb[2][row][m + k], s2);
    }
    h0[v] = s0; h1[v] = s1; h2[v] = s2;
  }
#endif

  // Phase 5: min-eigenvalue response; D layout: row = v + 8*hi, col = m
  float* rimg = resp + (size_t)b * HW;
#pragma unroll
  for (int v = 0; v < 8; ++v) {
    int row = v + 8 * hi;
    float A2 = h0[v], B2 = h1[v], AB = h2[v];
    float det = A2 * B2 - AB * AB;
    float tr = A2 + B2;
    float disc = fabsf(tr * tr - 4.f * det) + 1e-12f;
    rimg[(ty0 + row) * W + (tx0 + m)] = 0.5f * (tr - sqrtf(disc));
  }
}

// ---------------- K3: 5x5 NMS + 8x8 block-max -> candidate list -------------
__global__ __launch_bounds__(64) void k_nms_block(const float* __restrict__ resp,
                                                  float* __restrict__ candVal,
                                                  int* __restrict__ candIdx,
                                                  int* __restrict__ candCnt) {
  __shared__ float s[12][12];
  __shared__ float nvs[8][8];
  __shared__ float bmax;
  int b = blockIdx.z;
  int by = blockIdx.y * 8, bx = blockIdx.x * 8;
  int t = threadIdx.x, ty = t >> 3, tx = t & 7;
  const float* rimg = resp + (size_t)b * HW;
  for (int i = t; i < 144; i += 64) {
    int r = i / 12, c = i % 12;
    int gy = by - 2 + r, gx = bx - 2 + c;
    float v = -INFINITY;                 // reduce_window pads with -inf
    if (gy >= 0 && gy < H && gx >= 0 && gx < W) v = rimg[gy * W + gx];
    s[r][c] = v;
  }
  __syncthreads();
  float v = s[ty + 2][tx + 2];
  float mx = -INFINITY;
#pragma unroll
  for (int dy = 0; dy < 5; ++dy)
#pragma unroll
    for (int dx = 0; dx < 5; ++dx) mx = fmaxf(mx, s[ty + dy][tx + dx]);
  float nv = (v == mx) ? v : 0.f;
  nvs[ty][tx] = nv;
  __syncthreads();
  if (t == 0) {
    float bm = -INFINITY;
    for (int i = 0; i < 64; ++i) bm = fmaxf(bm, ((float*)nvs)[i]);
    bmax = bm;
  }
  __syncthreads();
  if (nv > 0.f && nv == bmax) {          // ties all emitted, matching reference
    int pos = atomicAdd(&candCnt[b], 1);
    if (pos < CAND_CAP) {
      candVal[b * CAND_CAP + pos] = nv;
      candIdx[b * CAND_CAP + pos] = (by + ty) * W + (bx + tx);
    }
  }
}

// ---------------- K4: per-image top-500 (bitonic) + mutual 5x5 NMS ----------
__global__ __launch_bounds__(1024) void k_select(const float* __restrict__ candVal,
                                                 const int* __restrict__ candIdx,
                                                 const int* __restrict__ candCnt,
                                                 int* __restrict__ cornIdx,
                                                 int* __restrict__ cornCnt) {
  __shared__ unsigned long long keys[NSORT];   // (desc value | asc index)
  int b = blockIdx.x, t = threadIdx.x;
  int cnt = candCnt[b];
  if (cnt > CAND_CAP) cnt = CAND_CAP;
  for (int i = t; i < NSORT; i += 1024) {
    unsigned long long k = 0xFFFFFFFFFFFFFFFFull;
    if (i < cnt) {
      unsigned bits = __float_as_uint(candVal[b * CAND_CAP + i]);
      unsigned ou = (bits & 0x80000000u) ? ~bits : (bits | 0x80000000u);
      unsigned inv = ~ou;                       // smaller inv == larger value
      k = ((unsigned long long)inv << 32) | (unsigned)candIdx[b * CAND_CAP + i];
    }
    keys[i] = k;
  }
  __syncthreads();
  for (int kk = 2; kk <= NSORT; kk <<= 1) {
    for (int j = kk >> 1; j > 0; j >>= 1) {
      for (int i = t; i < NSORT; i += 1024) {
        int ixj = i ^ j;
        if (ixj > i) {
          bool asc = ((i & kk) == 0);
          unsigned long long a = keys[i], c = keys[ixj];
          bool sw = asc ? (a > c) : (a < c);
          if (sw) { keys[i] = c; keys[ixj] = a; }
        }
      }
      __syncthreads();
    }
  }
  int n = cnt < NCORN ? cnt : NCORN;            // top_k(500); zeros never pass >0
  for (int i = t; i < n; i += 1024) {
    unsigned long long ki = keys[i];
    unsigned inv_i = (unsigned)(ki >> 32);
    int idx_i = (int)(ki & 0xFFFFFFFFull);
    int yi = idx_i / W, xi = idx_i % W;
    bool surv = true;
    for (int j = 0; j < n; ++j) {
      if (j == i) continue;
      unsigned long long kj = keys[j];
      unsigned inv_j = (unsigned)(kj >> 32);
      if (inv_j < inv_i) {                      // strictly greater value kills i
        int idx_j = (int)(kj & 0xFFFFFFFFull);
        int yj = idx_j / W, xj = idx_j % W;
        int dy = yj - yi, dx = xj - xi;
        if (dy <= 2 && dy >= -2 && dx <= 2 && dx >= -2) { surv = false; break; }
      }
    }
    if (surv) {
      int pos = atomicAdd(&cornCnt[b], 1);
      cornIdx[b * NCORN + pos] = idx_i;
    }
  }
}

// ---------------- K5: dense BCE base term: sum log1p(-p) --------------------
__global__ void k_base(const float* __restrict__ p, double* acc, int total) {
  __shared__ float red[256];
  int tid = threadIdx.x;
  int stride = gridDim.x * blockDim.x;
  float s = 0.f;
  for (int i = blockIdx.x * blockDim.x + tid; i < total; i += stride) {
    __builtin_prefetch(p + i + stride, 0, 0);   // global_prefetch_b8
    s += fmaxf(log1pf(-p[i]), -100.f);
  }
  red[tid] = s;
  __syncthreads();
  for (int w = 128; w > 0; w >>= 1) {
    if (tid < w) red[tid] += red[tid + w];
    __syncthreads();
  }
  if (tid == 0) atomicAdd(acc, (double)red[0]);
}

// ---------------- K6: sparse corner correction: sum (log p - log1p(-p)) -----
__global__ void k_corners(const float* __restrict__ p, const int* __restrict__ cornIdx,
                          const int* __restrict__ cornCnt, double* acc, int B) {
  __shared__ float red[256];
  int tid = threadIdx.x;
  float s = 0.f;
  for (int b = 0; b < B; ++b) {
    int cnt = cornCnt[b];
    if (cnt > NCORN) cnt = NCORN;
    for (int i = tid; i < cnt; i += 256) {
      float x = p[(size_t)b * HW + cornIdx[b * NCORN + i]];
      s += fmaxf(logf(x), -100.f) - fmaxf(log1pf(-x), -100.f);
    }
  }
  red[tid] = s;
  __syncthreads();
  for (int w = 128; w > 0; w >>= 1) {
    if (tid < w) red[tid] += red[tid + w];
    __syncthreads();
  }
  if (tid == 0) atomicAdd(acc, (double)red[0]);
}

// ---------------- K7: finalize loss = -S / N --------------------------------
__global__ void k_final(const double* acc, float* out, int total) {
  if (threadIdx.x == 0 && blockIdx.x == 0)
    out[0] = (float)(-acc[0] / (double)total);
}

extern "C" void kernel_launch(void* const* d_in, const int* in_sizes, int n_in,
                              void* d_out, int out_size, void* d_ws, size_t ws_size,
                              hipStream_t stream) {
  const float* scores = (const float*)d_in[0];
  const float* imgs = (const float*)d_in[1];
  const int B = in_sizes[0] / HW;
  const int totalPix = B * HW;

  char* ws = (char*)d_ws;
  size_t off = 0;
  auto alloc = [&](size_t bytes) -> void* {
    void* p = ws + off;
    off += bytes;
    off = (off + 255) & ~(size_t)255;
    return p;
  };
  float* gray = (float*)alloc((size_t)totalPix * 4);        // 16 MB
  float* resp = (float*)alloc((size_t)totalPix * 4);        // 16 MB
  float* candVal = (float*)alloc((size_t)B * CAND_CAP * 4);
  int* candIdx = (int*)alloc((size_t)B * CAND_CAP * 4);
  int* candCnt = (int*)alloc((size_t)B * 4);
  int* cornIdx = (int*)alloc((size_t)B * NCORN * 4);
  int* cornCnt = (int*)alloc((size_t)B * 4);
  double* acc = (double*)alloc(sizeof(double));
  (void)ws_size; (void)n_in; (void)out_size;

  k_init<<<1, 256, 0, stream>>>(acc, candCnt, cornCnt, B);
  k_gray<<<(totalPix + 255) / 256, 256, 0, stream>>>(imgs, gray, totalPix);
  k_gftt<<<dim3(W / 16, H / 16, B), 32, 0, stream>>>(gray, resp);
  k_nms_block<<<dim3(W / 8, H / 8, B), 64, 0, stream>>>(resp, candVal, candIdx, candCnt);
  k_select<<<B, 1024, 0, stream>>>(candVal, candIdx, candCnt, cornIdx, cornCnt);
  k_base<<<4096, 256, 0, stream>>>(scores, acc, totalPix);
  k_corners<<<1, 256, 0, stream>>>(scores, cornIdx, cornCnt, acc, B);
  k_final<<<1, 32, 0, stream>>>(acc, (float*)d_out, totalPix);
}